// FastBEV_export_model_635655160584
// MI455X (gfx1250) — compile-verified
//
#include <hip/hip_runtime.h>
#include <cstdint>
#include <cstddef>

// Problem geometry (compile-time constants from the reference)
#define N_SEQ   4
#define N_CAMS  6
#define C_CH    64
#define H_DIM   64
#define W_DIM   176
#define HW      (H_DIM * W_DIM)      // 11264 spatial positions per image
#define ROWS    (N_CAMS * HW)        // 67584 feature rows per sequence
#define ROWS_P  (ROWS + 1)           // +1 zero pad row (index == ROWS)
#define NVOX    160000               // 200*200*4 voxels

// ---- CDNA5 async global->LDS feature detection --------------------------
#if defined(__has_builtin)
#  if __has_builtin(__builtin_amdgcn_global_load_async_to_lds_b128)
#    define HAVE_ASYNC_LDS 1
#  endif
#  if __has_builtin(__builtin_amdgcn_s_wait_asynccnt)
#    define HAVE_WAIT_ASYNC_BUILTIN 1
#  endif
#endif

// Native clang vector types (HIP float4/int4 are structs, unusable with
// nontemporal/async builtins).
typedef float v4f __attribute__((ext_vector_type(4)));
typedef int   v4i __attribute__((vector_size(4 * sizeof(int))));
typedef __attribute__((address_space(1))) v4i* gptr_v4i;   // global
typedef __attribute__((address_space(3))) v4i* lptr_v4i;   // LDS

__device__ __forceinline__ void wait_asynccnt0() {
#if defined(HAVE_WAIT_ASYNC_BUILTIN)
  __builtin_amdgcn_s_wait_asynccnt(0);
#else
  asm volatile("s_wait_asynccnt 0" ::: "memory");
#endif
}

// -------------------------------------------------------------------------
// Pass 1: transpose features [img][C][S] -> ft[seq][cam*HW + s][C]
// so each gatherable row becomes a contiguous 256B chunk.
// Feature reads are non-temporal (read-once stream); ft stores are regular
// so the 69 MB table lands in (and stays in) the 192 MB L2.
// -------------------------------------------------------------------------
__global__ void __launch_bounds__(256)
transpose_feat(const float* __restrict__ f, float* __restrict__ ft) {
  __shared__ float tile[64][65];                    // pad 65: conflict-free
  const int img = blockIdx.y;                       // 0..23
  const int s0  = blockIdx.x * 64;                  // spatial tile origin
  const int tid = threadIdx.x;

  const float* src = f + (size_t)img * C_CH * HW;   // [c][s]
  // Load: 1024 float4 (= 64x64 floats), lanes sweep s fastest -> coalesced
  for (int i = 0; i < 4; ++i) {
    const int l4 = i * 256 + tid;
    const int c  = l4 >> 4;                         // 16 float4 per channel row
    const int s4 = l4 & 15;
    const v4f v =
        __builtin_nontemporal_load((const v4f*)(src + (size_t)c * HW + s0 + s4 * 4));
    tile[c][s4 * 4 + 0] = v[0];
    tile[c][s4 * 4 + 1] = v[1];
    tile[c][s4 * 4 + 2] = v[2];
    tile[c][s4 * 4 + 3] = v[3];
  }
  __syncthreads();

  const int seq = img / N_CAMS;
  const int cam = img % N_CAMS;
  float* dst = ft + ((size_t)seq * ROWS_P + (size_t)cam * HW + s0) * C_CH;
  // Store transposed: lanes sweep channel fastest -> coalesced 256B rows
  for (int i = 0; i < 4; ++i) {
    const int l4 = i * 256 + tid;
    const int s  = l4 >> 4;
    const int c4 = l4 & 15;
    v4f v;
    v[0] = tile[c4 * 4 + 0][s];
    v[1] = tile[c4 * 4 + 1][s];
    v[2] = tile[c4 * 4 + 2][s];
    v[3] = tile[c4 * 4 + 3][s];
    *(v4f*)(dst + (size_t)s * C_CH + c4 * 4) = v;   // RT: keep in L2
  }
}

// Write the zero pad row (row index == ROWS) for each sequence.
__global__ void __launch_bounds__(256)
write_pad(float* __restrict__ ft) {
  const int t   = threadIdx.x;      // 256 = 4 seq * 64 ch
  const int seq = t >> 6;
  const int c   = t & 63;
  ft[((size_t)seq * ROWS_P + ROWS) * C_CH + c] = 0.0f;
}

// -------------------------------------------------------------------------
// Pass 2: gather + transpose.
// Block = 256 threads, tile = 64 voxels.
//  1) 16 threads stage the 64 voxel indices into LDS (128b loads).
//  2) Each lane async-loads a 16B slice of one gathered 256B row straight
//     into the LDS tile (GLOBAL_LOAD_ASYNC_TO_LDS_B128, ASYNCcnt-tracked).
//  3) s_wait_asynccnt 0 + barrier, then transposed voxel-fastest coalesced
//     non-temporal stores (don't evict the L2-resident feature table).
// -------------------------------------------------------------------------
__global__ void __launch_bounds__(256)
gather_voxels(const float* __restrict__ ft, const int* __restrict__ xy,
              float* __restrict__ out) {
  __shared__ int   s_idx[64];
  __shared__ float vt[64][68];   // stride 68 floats: 16B-aligned rows, ~2-way banks
  const int seq = blockIdx.y;
  const int v0  = blockIdx.x * 64;
  const int tid = threadIdx.x;

  // Stage + clamp the tile's 64 indices once (instead of 16x per voxel).
  if (tid < 16) {
    v4i iv = ((const v4i*)(xy + (size_t)seq * NVOX + v0))[tid];
    for (int k = 0; k < 4; ++k) {
      int x = iv[k];
      x = x < 0 ? 0 : (x > ROWS ? ROWS : x);
      s_idx[tid * 4 + k] = x;
    }
  }
  __syncthreads();

  const float* base = ft + (size_t)seq * ROWS_P * C_CH;

  // 64 rows * 16 float4-slices = 1024 async B128 loads per tile (4 per lane)
  for (int i = 0; i < 4; ++i) {
    const int l4 = i * 256 + tid;
    const int v  = l4 >> 4;       // voxel within tile
    const int q  = l4 & 15;       // float4 slice within the 64-ch row
    const int idx = s_idx[v];     // LDS broadcast
    const float* g = base + (size_t)idx * C_CH + q * 4;
#if defined(HAVE_ASYNC_LDS)
    __builtin_amdgcn_global_load_async_to_lds_b128(
        (gptr_v4i)(__attribute__((address_space(1))) void*)(void*)g,
        (lptr_v4i)(__attribute__((address_space(3))) void*)&vt[v][q * 4],
        /*offset=*/0, /*cpol=*/0);
#else
    const v4f d = *(const v4f*)g;
    vt[v][q * 4 + 0] = d[0];
    vt[v][q * 4 + 1] = d[1];
    vt[v][q * 4 + 2] = d[2];
    vt[v][q * 4 + 3] = d[3];
#endif
  }
#if defined(HAVE_ASYNC_LDS)
  wait_asynccnt0();               // my async LDS writes have landed
#endif
  __syncthreads();                // everyone's tile slice visible

  // Transposed write-out: out[(seq*64+c)*NVOX + v0 + v], voxel fastest.
  float* o = out + (size_t)seq * C_CH * NVOX + v0;
  for (int i = 0; i < 16; ++i) {
    const int l = i * 256 + tid;
    const int v = l & 63;
    const int c = l >> 6;
    __builtin_nontemporal_store(vt[v][c], o + (size_t)c * NVOX + v);
  }
}

// -------------------------------------------------------------------------
// Fallback (only if workspace is too small): direct strided gather.
// -------------------------------------------------------------------------
__global__ void __launch_bounds__(256)
gather_direct(const float* __restrict__ f, const int* __restrict__ xy,
              float* __restrict__ out, size_t total) {
  const size_t t = (size_t)blockIdx.x * blockDim.x + threadIdx.x;
  if (t >= total) return;
  const int    v   = (int)(t % NVOX);
  const size_t r   = t / NVOX;
  const int    c   = (int)(r % C_CH);
  const int    seq = (int)(r / C_CH);
  const int idx = xy[(size_t)seq * NVOX + v];
  float val = 0.0f;
  if (idx >= 0 && idx < ROWS) {
    const int cam = idx / HW;
    const int s   = idx % HW;
    val = f[(((size_t)(seq * N_CAMS + cam)) * C_CH + c) * HW + s];
  }
  __builtin_nontemporal_store(val, out + t);
}

// -------------------------------------------------------------------------
extern "C" void kernel_launch(void* const* d_in, const int* in_sizes, int n_in,
                              void* d_out, int out_size, void* d_ws, size_t ws_size,
                              hipStream_t stream) {
  (void)in_sizes; (void)n_in; (void)out_size;
  const float* f   = (const float*)d_in[0];   // [24, 64, 64, 176] f32
  const int*   xy  = (const int*)d_in[1];     // [4, 160000] i32 in [0, ROWS]
  float*       out = (float*)d_out;           // [1, 256, 200, 200, 4] f32

  const size_t need = (size_t)N_SEQ * ROWS_P * C_CH * sizeof(float); // ~69 MB
  if (ws_size >= need) {
    float* ft = (float*)d_ws;
    transpose_feat<<<dim3(HW / 64, N_SEQ * N_CAMS), 256, 0, stream>>>(f, ft);
    write_pad<<<1, 256, 0, stream>>>(ft);
    gather_voxels<<<dim3(NVOX / 64, N_SEQ), 256, 0, stream>>>(ft, xy, out);
  } else {
    const size_t total = (size_t)N_SEQ * C_CH * NVOX;
    gather_direct<<<(unsigned)((total + 255) / 256), 256, 0, stream>>>(f, xy, out, total);
  }
}